// EFC_8778913153624
// MI455X (gfx1250) — compile-verified
//
#include <hip/hip_runtime.h>
#include <cstdint>

#define N_   8
#define C_   256
#define H_   80
#define W_   80
#define HW_  6400
#define G_   16
#define CPG_ 16
#define BIG_ 13107200      // N_*C_*HW_

typedef float v2f __attribute__((ext_vector_type(2)));
typedef float v8f __attribute__((ext_vector_type(8)));
typedef unsigned int u32x4 __attribute__((ext_vector_type(4)));
typedef int i32x4 __attribute__((ext_vector_type(4)));
typedef int i32x8 __attribute__((ext_vector_type(8)));

__device__ __forceinline__ float sigmoidf_(float x) { return 1.0f / (1.0f + __expf(-x)); }

// ---------------------------------------------------------------------------
// Stage a contiguous 4096-float (16 KB) weight slab into LDS.
// Primary path: Tensor Data Mover (tensor_load_to_lds, 6-arg clang-23 form)
// issued once by wave 0, completion via s_wait_tensorcnt.
// Fallback: plain cooperative copy. Caller must __syncthreads() afterwards.
// ---------------------------------------------------------------------------
__device__ __forceinline__ void tdm_load_slab(const float* gsrc, float* lds_dst)
{
#if __has_builtin(__builtin_amdgcn_tensor_load_to_lds) && __has_builtin(__builtin_amdgcn_s_wait_tensorcnt)
  if ((threadIdx.x >> 5) == 0) {            // one wave issues the DMA (TDM ignores EXEC)
    uint64_t ga = (uint64_t)(uintptr_t)gsrc;
    uint32_t lo = (uint32_t)(uintptr_t)lds_dst;   // low 32b of generic ptr == LDS byte offset
    u32x4 g0;
    g0[0] = 1u;                                          // count=1, user descriptor
    g0[1] = lo;                                          // lds_addr (bytes)
    g0[2] = (uint32_t)(ga & 0xFFFFFFFFu);                // global_addr[31:0]
    g0[3] = (uint32_t)((ga >> 32) & 0x01FFFFFFu) | (2u << 30);  // addr[56:32] | type=2
    i32x8 g1;
    g1[0] = (int)(2u << 16);                 // data_size = 4 bytes
    g1[1] = (int)((4096u & 0xFFFFu) << 16);  // tensor_dim0[15:0]  (bits 63:48)
    g1[2] = (int)((4096u >> 16) | (1u << 16)); // tensor_dim0[31:16] | tensor_dim1=1
    g1[3] = (int)(4096u << 16);              // tile_dim0 = 4096   (bits 127:112)
    g1[4] = 0;                               // tile_dim1 = 0 (1-D), tile_dim2 = 0
    g1[5] = 4096;                            // tensor_dim0_stride
    g1[6] = 0; g1[7] = 0;
    i32x4 z4; z4[0] = 0; z4[1] = 0; z4[2] = 0; z4[3] = 0;
    i32x8 z8; z8[0] = 0; z8[1] = 0; z8[2] = 0; z8[3] = 0;
    z8[4] = 0; z8[5] = 0; z8[6] = 0; z8[7] = 0;
    __builtin_amdgcn_tensor_load_to_lds(g0, g1, z4, z4, z8, 0);
    __builtin_amdgcn_s_wait_tensorcnt(0);
  }
#else
  for (int i = threadIdx.x; i < 4096; i += blockDim.x) lds_dst[i] = gsrc[i];
#endif
}

// ---------------------------------------------------------------------------
// Kernel 1: g1 = w1@x1 + b1 ; g2 = w2@x2 + b2   (WMMA f32 16x16x4)
// grid(25,16,8) block(128): wave tile = 16 oc x 64 px, block = 16 oc x 256 px
// ---------------------------------------------------------------------------
__global__ __launch_bounds__(128)
void k_gemm12(const float* __restrict__ x1, const float* __restrict__ x2,
              const float* __restrict__ w1, const float* __restrict__ b1,
              const float* __restrict__ w2, const float* __restrict__ b2,
              float* __restrict__ g1o, float* __restrict__ g2o)
{
  __shared__ float sw1[16 * 256];
  __shared__ float sw2[16 * 256];
  const int n   = blockIdx.z;
  const int oc0 = blockIdx.y * 16;
  tdm_load_slab(w1 + (size_t)oc0 * C_, sw1);
  tdm_load_slab(w2 + (size_t)oc0 * C_, sw2);
  __syncthreads();

  const int wave  = threadIdx.x >> 5;
  const int lane  = threadIdx.x & 31;
  const int lp    = lane & 15;
  const int khalf = (lane >> 4) << 1;     // 0 or 2
  const int pix0  = blockIdx.x * 256 + wave * 64;

  const float* xb1 = x1 + (size_t)n * C_ * HW_;
  const float* xb2 = x2 + (size_t)n * C_ * HW_;

  v8f acc1[4], acc2[4];
#pragma unroll
  for (int s = 0; s < 4; ++s)
#pragma unroll
    for (int j = 0; j < 8; ++j) { acc1[s][j] = 0.f; acc2[s][j] = 0.f; }

  for (int k0 = 0; k0 < C_; k0 += 4) {
    const int kk = k0 + khalf;
    v2f a1 = *(const v2f*)&sw1[lp * 256 + kk];
    v2f a2 = *(const v2f*)&sw2[lp * 256 + kk];
#pragma unroll
    for (int s = 0; s < 4; ++s) {
      const int pix = pix0 + s * 16 + lp;
      v2f bb1, bb2;
      bb1.x = xb1[(size_t)kk * HW_ + pix];
      bb1.y = xb1[(size_t)(kk + 1) * HW_ + pix];
      bb2.x = xb2[(size_t)kk * HW_ + pix];
      bb2.y = xb2[(size_t)(kk + 1) * HW_ + pix];
      acc1[s] = __builtin_amdgcn_wmma_f32_16x16x4_f32(false, a1, false, bb1, (short)0, acc1[s], false, false);
      acc2[s] = __builtin_amdgcn_wmma_f32_16x16x4_f32(false, a2, false, bb2, (short)0, acc2[s], false, false);
    }
  }

#pragma unroll
  for (int s = 0; s < 4; ++s) {
    const int pix = pix0 + s * 16 + lp;
#pragma unroll
    for (int j = 0; j < 8; ++j) {
      const int oc = oc0 + j + ((lane >> 4) << 3);
      const size_t idx = ((size_t)n * C_ + oc) * HW_ + pix;
      g1o[idx] = acc1[s][j] + b1[oc];
      g2o[idx] = acc2[s][j] + b2[oc];
    }
  }
}

// ---------------------------------------------------------------------------
// Kernel 2: per-(n,c) sum / sumsq of XG = g1+g2  (block per n*C+c)
// ---------------------------------------------------------------------------
__global__ __launch_bounds__(256)
void k_chanstats(const float* __restrict__ g1, const float* __restrict__ g2,
                 float* __restrict__ csum, float* __restrict__ csum2)
{
  const int nc = blockIdx.x;
  const float* p1 = g1 + (size_t)nc * HW_;
  const float* p2 = g2 + (size_t)nc * HW_;
  float s = 0.f, s2 = 0.f;
  for (int i = threadIdx.x; i < HW_; i += 256) {
    float v = p1[i] + p2[i];
    s += v; s2 += v * v;
  }
  __shared__ float rs[256], rs2[256];
  const int t = threadIdx.x;
  rs[t] = s; rs2[t] = s2; __syncthreads();
  for (int off = 128; off > 0; off >>= 1) {
    if (t < off) { rs[t] += rs[t + off]; rs2[t] += rs2[t + off]; }
    __syncthreads();
  }
  if (t == 0) { csum[nc] = rs[0]; csum2[nc] = rs2[0]; }
}

// ---------------------------------------------------------------------------
// Kernel 3: rw = sigmoid(mean); group mean / 1/(unbiased std + eps)
// ---------------------------------------------------------------------------
__global__ __launch_bounds__(256)
void k_finstats(const float* __restrict__ csum, const float* __restrict__ csum2,
                float* __restrict__ rw, float* __restrict__ gmean, float* __restrict__ ginv)
{
  const int t = blockIdx.x * 256 + threadIdx.x;
  if (t < N_ * C_) rw[t] = sigmoidf_(csum[t] * (1.0f / HW_));
  if (t < N_ * G_) {
    const int n = t / G_, g = t % G_;
    float S = 0.f, SS = 0.f;
    for (int c = g * CPG_; c < (g + 1) * CPG_; ++c) { S += csum[n * C_ + c]; SS += csum2[n * C_ + c]; }
    const float M = (float)(CPG_ * HW_);
    const float mean = S / M;
    const float var = (SS - S * S / M) / (M - 1.0f);
    const float sd = sqrtf(fmaxf(var, 0.f));
    gmean[t] = mean;
    ginv[t] = 1.0f / (sd + 1e-10f);
  }
}

// ---------------------------------------------------------------------------
// Kernel 4: per-pixel gate4 = sigmoid(wg . XG + bg)
// ---------------------------------------------------------------------------
__global__ __launch_bounds__(256)
void k_gate(const float* __restrict__ g1, const float* __restrict__ g2,
            const float* __restrict__ wg, const float* __restrict__ bg,
            float* __restrict__ gate4)
{
  const int t = blockIdx.x * 256 + threadIdx.x;
  if (t >= N_ * HW_) return;
  const int n = t / HW_, pix = t % HW_;
  const float* p1 = g1 + (size_t)n * C_ * HW_ + pix;
  const float* p2 = g2 + (size_t)n * C_ * HW_ + pix;
  float acc = 0.f;
  for (int c = 0; c < C_; ++c)
    acc += wg[c] * (p1[(size_t)c * HW_] + p2[(size_t)c * HW_]);
  gate4[t] = sigmoidf_(acc + bg[0]);
}

// ---------------------------------------------------------------------------
// Kernel 5: masks -> xlow, xup, xga = gate4*XG ; gm = mean(xlow)  (block per n,c)
// ---------------------------------------------------------------------------
__global__ __launch_bounds__(256)
void k_masks(const float* __restrict__ g1, const float* __restrict__ g2,
             const float* __restrict__ rw, const float* __restrict__ gate4,
             const float* __restrict__ bnw, const float* __restrict__ bnb,
             const float* __restrict__ bnrm, const float* __restrict__ bnrv,
             float* __restrict__ xlow, float* __restrict__ xup,
             float* __restrict__ xga, float* __restrict__ gm)
{
  const int nc = blockIdx.x;
  const int n = nc / C_, c = nc % C_;
  const float scale = bnw[c] * rsqrtf(bnrv[c] + 1e-5f);
  const float shift = bnb[c] - bnrm[c] * scale;
  const float rwv = rw[nc];
  const float* p1 = g1 + (size_t)nc * HW_;
  const float* p2 = g2 + (size_t)nc * HW_;
  const float* gp = gate4 + (size_t)n * HW_;
  float s = 0.f;
  for (int i = threadIdx.x; i < HW_; i += 256) {
    const float a = p1[i], b = p2[i];
    const float X = a + b;
    const float w1e = sigmoidf_(a * scale + shift);
    const float w2e = sigmoidf_(b * scale + shift);
    const float up = (rwv >= w1e ? 1.f : 0.f) + (rwv >= w2e ? 1.f : 0.f);
    const float lo = (rwv <  w1e ? 1.f : 0.f) + (rwv <  w2e ? 1.f : 0.f);
    const float xl = lo * X;
    const size_t idx = (size_t)nc * HW_ + i;
    xlow[idx] = xl;
    xup[idx]  = up * X;
    xga[idx]  = gp[i] * X;
    s += xl;
  }
  __shared__ float rs[256];
  const int t = threadIdx.x;
  rs[t] = s; __syncthreads();
  for (int off = 128; off > 0; off >>= 1) {
    if (t < off) rs[t] += rs[t + off];
    __syncthreads();
  }
  if (t == 0) gm[nc] = rs[0] * (1.0f / HW_);
}

// ---------------------------------------------------------------------------
// Kernel 6: x_so = softmax_c(relu(wgate @ gm + bgate))   (block per n)
// ---------------------------------------------------------------------------
__global__ __launch_bounds__(256)
void k_xso(const float* __restrict__ gm, const float* __restrict__ wgate,
           const float* __restrict__ bgate, float* __restrict__ xso)
{
  const int n = blockIdx.x, c = threadIdx.x;
  __shared__ float sg[256];
  __shared__ float red[256];
  sg[c] = gm[n * C_ + c];
  __syncthreads();
  float t = 0.f;
  for (int k = 0; k < C_; ++k) t += wgate[c * C_ + k] * sg[k];
  t = fmaxf(t + bgate[c], 0.f);
  red[c] = t; __syncthreads();
  for (int off = 128; off > 0; off >>= 1) {
    if (c < off) red[c] = fmaxf(red[c], red[c + off]);
    __syncthreads();
  }
  const float mx = red[0]; __syncthreads();
  const float e = __expf(t - mx);
  red[c] = e; __syncthreads();
  for (int off = 128; off > 0; off >>= 1) {
    if (c < off) red[c] += red[c + off];
    __syncthreads();
  }
  xso[n * C_ + c] = e / red[0];
}

// ---------------------------------------------------------------------------
// Kernel 7: depthwise 3x3 SAME on xlow -> dwt
// ---------------------------------------------------------------------------
__global__ __launch_bounds__(256)
void k_dw(const float* __restrict__ xlow, const float* __restrict__ wdw,
          const float* __restrict__ bdw, float* __restrict__ dwt)
{
  const size_t t = (size_t)blockIdx.x * 256 + threadIdx.x;
  if (t >= (size_t)BIG_) return;
  const int pix = (int)(t % HW_);
  const int nc  = (int)(t / HW_);
  const int c   = nc % C_;
  const int h = pix / W_, w = pix % W_;
  const float* base = xlow + (size_t)nc * HW_;
  float acc = bdw[c];
#pragma unroll
  for (int dy = -1; dy <= 1; ++dy)
#pragma unroll
    for (int dx = -1; dx <= 1; ++dx) {
      const int hh = h + dy, ww = w + dx;
      if (hh >= 0 && hh < H_ && ww >= 0 && ww < W_)
        acc += wdw[c * 9 + (dy + 1) * 3 + (dx + 1)] * base[hh * W_ + ww];
    }
  dwt[t] = acc;
}

// ---------------------------------------------------------------------------
// Kernel 8: y3 = w3@dwt + b3 ; y4 = w4@xup + b4 ;
//           out = y3*xso + y4 + ((xga - mean)/(std+eps))*gamma + beta
// ---------------------------------------------------------------------------
__global__ __launch_bounds__(128)
void k_final(const float* __restrict__ dwt, const float* __restrict__ xup,
             const float* __restrict__ xga,
             const float* __restrict__ w3, const float* __restrict__ b3,
             const float* __restrict__ w4, const float* __restrict__ b4,
             const float* __restrict__ xso, const float* __restrict__ gmean,
             const float* __restrict__ ginv, const float* __restrict__ gamma,
             const float* __restrict__ beta, float* __restrict__ out)
{
  __shared__ float sw3[16 * 256];
  __shared__ float sw4[16 * 256];
  const int n   = blockIdx.z;
  const int oc0 = blockIdx.y * 16;
  tdm_load_slab(w3 + (size_t)oc0 * C_, sw3);
  tdm_load_slab(w4 + (size_t)oc0 * C_, sw4);
  __syncthreads();

  const int wave  = threadIdx.x >> 5;
  const int lane  = threadIdx.x & 31;
  const int lp    = lane & 15;
  const int khalf = (lane >> 4) << 1;
  const int pix0  = blockIdx.x * 256 + wave * 64;

  const float* xbA = dwt + (size_t)n * C_ * HW_;
  const float* xbB = xup + (size_t)n * C_ * HW_;

  v8f acc1[4], acc2[4];
#pragma unroll
  for (int s = 0; s < 4; ++s)
#pragma unroll
    for (int j = 0; j < 8; ++j) { acc1[s][j] = 0.f; acc2[s][j] = 0.f; }

  for (int k0 = 0; k0 < C_; k0 += 4) {
    const int kk = k0 + khalf;
    v2f a1 = *(const v2f*)&sw3[lp * 256 + kk];
    v2f a2 = *(const v2f*)&sw4[lp * 256 + kk];
#pragma unroll
    for (int s = 0; s < 4; ++s) {
      const int pix = pix0 + s * 16 + lp;
      v2f bb1, bb2;
      bb1.x = xbA[(size_t)kk * HW_ + pix];
      bb1.y = xbA[(size_t)(kk + 1) * HW_ + pix];
      bb2.x = xbB[(size_t)kk * HW_ + pix];
      bb2.y = xbB[(size_t)(kk + 1) * HW_ + pix];
      acc1[s] = __builtin_amdgcn_wmma_f32_16x16x4_f32(false, a1, false, bb1, (short)0, acc1[s], false, false);
      acc2[s] = __builtin_amdgcn_wmma_f32_16x16x4_f32(false, a2, false, bb2, (short)0, acc2[s], false, false);
    }
  }

#pragma unroll
  for (int s = 0; s < 4; ++s) {
    const int pix = pix0 + s * 16 + lp;
#pragma unroll
    for (int j = 0; j < 8; ++j) {
      const int oc = oc0 + j + ((lane >> 4) << 3);
      const size_t idx = ((size_t)n * C_ + oc) * HW_ + pix;
      const int g = oc >> 4;
      const float y3 = acc1[s][j] + b3[oc];
      const float y4 = acc2[s][j] + b4[oc];
      const float gn = (xga[idx] - gmean[n * G_ + g]) * ginv[n * G_ + g] * gamma[oc] + beta[oc];
      out[idx] = y3 * xso[n * C_ + oc] + y4 + gn;
    }
  }
}

// ---------------------------------------------------------------------------
extern "C" void kernel_launch(void* const* d_in, const int* in_sizes, int n_in,
                              void* d_out, int out_size, void* d_ws, size_t ws_size,
                              hipStream_t stream)
{
  const float* x1    = (const float*)d_in[0];
  const float* x2    = (const float*)d_in[1];
  const float* w1    = (const float*)d_in[2];
  const float* b1    = (const float*)d_in[3];
  const float* w2    = (const float*)d_in[4];
  const float* b2    = (const float*)d_in[5];
  const float* w3    = (const float*)d_in[6];
  const float* b3    = (const float*)d_in[7];
  const float* w4    = (const float*)d_in[8];
  const float* b4    = (const float*)d_in[9];
  const float* wg    = (const float*)d_in[10];
  const float* bg    = (const float*)d_in[11];
  const float* wdw   = (const float*)d_in[12];
  const float* bdw   = (const float*)d_in[13];
  // d_in[14] (wi), d_in[15] (bi): dead code in reference (softmax over singleton axis == 1)
  const float* wgate = (const float*)d_in[16];
  const float* bgate = (const float*)d_in[17];
  const float* bnw   = (const float*)d_in[18];
  const float* bnb   = (const float*)d_in[19];
  const float* bnrm  = (const float*)d_in[20];
  const float* bnrv  = (const float*)d_in[21];
  const float* gamma = (const float*)d_in[22];
  const float* beta  = (const float*)d_in[23];

  float* ws   = (float*)d_ws;
  float* g1   = ws;                       // BIG_
  float* g2   = ws + (size_t)BIG_;        // BIG_
  float* xlow = ws + (size_t)2 * BIG_;    // BIG_
  float* xup  = ws + (size_t)3 * BIG_;    // BIG_
  float* xga  = ws + (size_t)4 * BIG_;    // BIG_
  float* dwt  = g1;                       // alias: g1 is dead after k_masks/k_gate
  float* sm   = ws + (size_t)5 * BIG_;
  float* csum  = sm;           // 2048
  float* csum2 = sm + 2048;    // 2048
  float* rw    = sm + 4096;    // 2048
  float* gmean = sm + 6144;    // 128
  float* ginv  = sm + 6272;    // 128
  float* gate4 = sm + 6400;    // 51200
  float* gm    = sm + 57600;   // 2048
  float* xso   = sm + 59648;   // 2048

  k_gemm12 <<<dim3(25, 16, 8), 128, 0, stream>>>(x1, x2, w1, b1, w2, b2, g1, g2);
  k_chanstats<<<N_ * C_, 256, 0, stream>>>(g1, g2, csum, csum2);
  k_finstats <<<8, 256, 0, stream>>>(csum, csum2, rw, gmean, ginv);
  k_gate     <<<(N_ * HW_) / 256, 256, 0, stream>>>(g1, g2, wg, bg, gate4);
  k_masks    <<<N_ * C_, 256, 0, stream>>>(g1, g2, rw, gate4, bnw, bnb, bnrm, bnrv,
                                           xlow, xup, xga, gm);
  k_xso      <<<N_, 256, 0, stream>>>(gm, wgate, bgate, xso);
  k_dw       <<<BIG_ / 256, 256, 0, stream>>>(xlow, wdw, bdw, dwt);
  k_final    <<<dim3(25, 16, 8), 128, 0, stream>>>(dwt, xup, xga, w3, b3, w4, b4,
                                                   xso, gmean, ginv, gamma, beta,
                                                   (float*)d_out);
  (void)in_sizes; (void)n_in; (void)out_size; (void)ws_size;
}